// VEGS_60644938219817
// MI455X (gfx1250) — compile-verified
//
#include <hip/hip_runtime.h>
#include <hip/hip_bf16.h>

#define NG 32768
#define MP 8192
#define KF 12            // padded feature rank (10 real + 2 zero)
#define GSPLIT 4         // gaussian-range split for extra parallelism
#define BG_VAL (-1.0f)

// fold exp(-0.5*q) = exp2(q * (-0.5*log2 e)) into the gaussian coefficients
#define SCL (-0.72134752044448170368f)   // -0.5 * log2(e)

typedef float v2f __attribute__((ext_vector_type(2)));
typedef float v8f __attribute__((ext_vector_type(8)));

#if __has_builtin(__builtin_amdgcn_wmma_f32_16x16x4_f32)
#define HAVE_WMMA_F32X4 1
#else
#define HAVE_WMMA_F32X4 0
#endif

__device__ __forceinline__ float fast_exp2(float x) {
#if __has_builtin(__builtin_amdgcn_exp2f)
    return __builtin_amdgcn_exp2f(x);
#else
    return exp2f(x);
#endif
}

// ---------------------------------------------------------------------------
// Kernel 1: per-gaussian coefficients, K-PAIRED layout:
//   coefP[(k>>1)*(2*NG) + 2*g + (k&1)]  -> one b64 load per fragment half.
//   All 10 live coefficients pre-scaled by SCL so the WMMA output feeds
//   v_exp_f32 (2^x) directly with no per-element multiplies.
// ---------------------------------------------------------------------------
__global__ __launch_bounds__(256) void precompute_gauss(
    const float* __restrict__ xyz, const float* __restrict__ wraw,
    const float* __restrict__ sraw, const float* __restrict__ rraw,
    const float* __restrict__ vraw, float* __restrict__ coefP,
    float* __restrict__ vval)
{
    int g = blockIdx.x * 256 + threadIdx.x;
    if (g >= NG) return;

    float sx = __expf(sraw[g*3+0]);
    float sy = __expf(sraw[g*3+1]);
    float sz = __expf(sraw[g*3+2]);
    float rr = sqrtf(sx*sx + sy*sy + sz*sz) + 1e-8f;
    float rsoft = 0.02f * tanhf(rr * (1.0f/0.02f));
    float sc = rsoft / rr;
    sx *= sc; sy *= sc; sz *= sc;
    float i0 = 1.0f/(sx*sx), i1 = 1.0f/(sy*sy), i2 = 1.0f/(sz*sz);

    float qw = rraw[g*4+0], qx = rraw[g*4+1], qy = rraw[g*4+2], qz = rraw[g*4+3];
    float qn = 1.0f/(sqrtf(qw*qw+qx*qx+qy*qy+qz*qz) + 1e-12f);
    qw*=qn; qx*=qn; qy*=qn; qz*=qn;
    float r00 = 1.f-2.f*(qy*qy+qz*qz), r01 = 2.f*(qx*qy-qw*qz), r02 = 2.f*(qx*qz+qw*qy);
    float r10 = 2.f*(qx*qy+qw*qz),     r11 = 1.f-2.f*(qx*qx+qz*qz), r12 = 2.f*(qy*qz-qw*qx);
    float r20 = 2.f*(qx*qz-qw*qy),     r21 = 2.f*(qy*qz+qw*qx), r22 = 1.f-2.f*(qx*qx+qy*qy);

    // A = R * diag(1/s^2) * R^T (symmetric)
    float A00 = r00*r00*i0 + r01*r01*i1 + r02*r02*i2;
    float A01 = r00*r10*i0 + r01*r11*i1 + r02*r12*i2;
    float A02 = r00*r20*i0 + r01*r21*i1 + r02*r22*i2;
    float A11 = r10*r10*i0 + r11*r11*i1 + r12*r12*i2;
    float A12 = r10*r20*i0 + r11*r21*i1 + r12*r22*i2;
    float A22 = r20*r20*i0 + r21*r21*i1 + r22*r22*i2;

    float mx = xyz[g*3+0], my = xyz[g*3+1], mz = xyz[g*3+2];
    float b0 = -2.f*(A00*mx + A01*my + A02*mz);
    float b1 = -2.f*(A01*mx + A11*my + A12*mz);
    float b2 = -2.f*(A02*mx + A12*my + A22*mz);
    float muAmu = A00*mx*mx + A11*my*my + A22*mz*mz
                + 2.f*(A01*mx*my + A02*mx*mz + A12*my*mz);

    float w = 1.0f/(1.0f + __expf(-wraw[g]));
    float cst = muAmu - 2.0f*__logf(w);

    float* c = coefP + 2*g;
    const int S = 2*NG;
    c[0*S+0] = SCL*A00;      c[0*S+1] = SCL*A11;        // k0,k1
    c[1*S+0] = SCL*A22;      c[1*S+1] = SCL*2.f*A01;    // k2,k3
    c[2*S+0] = SCL*2.f*A02;  c[2*S+1] = SCL*2.f*A12;    // k4,k5
    c[3*S+0] = SCL*b0;       c[3*S+1] = SCL*b1;         // k6,k7
    c[4*S+0] = SCL*b2;       c[4*S+1] = SCL*cst;        // k8,k9
    c[5*S+0] = 0.f;          c[5*S+1] = 0.f;            // k10,k11 pad

    vval[g] = 1.0f/(1.0f + __expf(-vraw[g]));
}

// ---------------------------------------------------------------------------
// Kernel 2: per-point features, same K-paired layout.
// ---------------------------------------------------------------------------
__global__ __launch_bounds__(256) void precompute_points(
    const float* __restrict__ x, float* __restrict__ featP)
{
    int p = blockIdx.x * 256 + threadIdx.x;
    if (p >= MP) return;
    float px = (x[p*3+0] + 1.0f) * 0.5f;
    float py = (x[p*3+1] + 1.0f) * 0.5f;
    float pz = (x[p*3+2] + 1.0f) * 0.5f;
    float* f = featP + 2*p;
    const int S = 2*MP;
    f[0*S+0] = px*px;  f[0*S+1] = py*py;
    f[1*S+0] = pz*pz;  f[1*S+1] = px*py;
    f[2*S+0] = px*pz;  f[2*S+1] = py*pz;
    f[3*S+0] = px;     f[3*S+1] = py;
    f[4*S+0] = pz;     f[4*S+1] = 1.0f;
    f[5*S+0] = 0.f;    f[5*S+1] = 0.f;
}

#if HAVE_WMMA_F32X4
// ---------------------------------------------------------------------------
// Kernel 3 (WMMA path): grid = (MP/16 point tiles, GSPLIT gaussian chunks),
// 8 waves per block; each wave runs a COMPILE-TIME-CONSTANT 64-iteration loop
// (scalar s_cmp/s_cbranch control, EXEC stays all-ones around the WMMAs) with
// pointer-bumped fragment loads.
//
// Fragment layouts per CDNA5 ISA (wave32):
//   A 16x4 f32: lane holds row M=lane%16; elem j -> K = 2*(lane/16)+j
//   B 4x16 f32: lane holds col N=lane%16; elem j -> K = 2*(lane/16)+j
//   C/D 16x16 f32: elem r -> (M = r + 8*(lane/16), N = lane%16)
// ---------------------------------------------------------------------------
__global__ __launch_bounds__(256) void splat_main(
    const float* __restrict__ featP,  // paired [KF/2][MP][2]
    const float* __restrict__ coefP,  // paired [KF/2][NG][2]
    const float* __restrict__ vval,   // [NG]
    float* __restrict__ numP,         // [GSPLIT][MP]
    float* __restrict__ denP)         // [GSPLIT][MP]
{
    __shared__ float lds_num[8][16];
    __shared__ float lds_den[8][16];

    const int tid  = threadIdx.x;
    const int wave = tid >> 5;
    const int lane = tid & 31;
    const int half = lane >> 4;      // which K-pair this half-wave carries
    const int l16  = lane & 15;
    const int pm   = blockIdx.x * 16 + l16;   // point row for A fragment
    const int gy   = blockIdx.y;              // gaussian chunk

    // Loop-invariant A fragments: K-pairs {0,1},{2,3} / {4,5},{6,7} / {8,9},{10,11}
    const v2f a0 = *(const v2f*)(featP + (0 + half)*(2*MP) + 2*pm);
    const v2f a1 = *(const v2f*)(featP + (2 + half)*(2*MP) + 2*pm);
    const v2f a2 = *(const v2f*)(featP + (4 + half)*(2*MP) + 2*pm);

    const int NT    = NG / 16;        // 2048 gaussian tiles total
    const int NTC   = NT / GSPLIT;    // 512 tiles per chunk
    const int NITER = NTC / 8;        // 64 iterations per wave (compile-time)

    // This wave's first gaussian column; stride 8 tiles = 128 gaussians / iter.
    const int gn0 = (gy * NTC + wave) * 16 + l16;
    const float* bp  = coefP + (half)*(2*NG) + 2*gn0;  // +4*NG, +8*NG for b1,b2
    const float* vvp = vval + gn0;

    float num[8], den[8];
#pragma unroll
    for (int r = 0; r < 8; ++r) { num[r] = 0.f; den[r] = 0.f; }

#pragma unroll 2
    for (int it = 0; it < NITER; ++it) {
        const v2f b0 = *(const v2f*)(bp);
        const v2f b1 = *(const v2f*)(bp + 4*NG);
        const v2f b2 = *(const v2f*)(bp + 8*NG);
        const float vv = *vvp;
        bp  += 8 * 16 * 2;     // 8 tiles * 16 cols * 2 floats
        vvp += 8 * 16;

        v8f c = {};
        c = __builtin_amdgcn_wmma_f32_16x16x4_f32(false, a0, false, b0,
                                                  (short)0, c, false, false);
        c = __builtin_amdgcn_wmma_f32_16x16x4_f32(false, a1, false, b1,
                                                  (short)0, c, false, false);
        c = __builtin_amdgcn_wmma_f32_16x16x4_f32(false, a2, false, b2,
                                                  (short)0, c, false, false);

#pragma unroll
        for (int r = 0; r < 8; ++r) {
            float alpha = fast_exp2(c[r]);   // -0.5*log2e and weight pre-folded
            num[r] += alpha * vv;
            den[r] += alpha;
        }
    }

    // Reduce over N (gaussian columns) = across the 16 lanes of each half.
#pragma unroll
    for (int r = 0; r < 8; ++r) {
        float n = num[r], d = den[r];
        for (int m = 1; m < 16; m <<= 1) {
            n += __shfl_xor(n, m, 32);
            d += __shfl_xor(d, m, 32);
        }
        if (l16 == 0) {                       // lane 0 -> M=r, lane 16 -> M=8+r
            lds_num[wave][half*8 + r] = n;
            lds_den[wave][half*8 + r] = d;
        }
    }
    __syncthreads();

    if (tid < 16) {
        float n = 0.f, d = 0.f;
        for (int wv = 0; wv < 8; ++wv) { n += lds_num[wv][tid]; d += lds_den[wv][tid]; }
        const int p = blockIdx.x * 16 + tid;
        numP[gy*MP + p] = n;
        denP[gy*MP + p] = d;
    }
}

// Kernel 4: sum the GSPLIT partials and apply the background predicate.
__global__ __launch_bounds__(256) void finalize(
    const float* __restrict__ numP, const float* __restrict__ denP,
    float* __restrict__ out)
{
    int p = blockIdx.x * 256 + threadIdx.x;
    if (p >= MP) return;
    float n = 0.f, d = 0.f;
#pragma unroll
    for (int gy = 0; gy < GSPLIT; ++gy) { n += numP[gy*MP + p]; d += denP[gy*MP + p]; }
    out[p] = (d > 1e-8f) ? n : BG_VAL;
}
#else
// Fallback (no f32x4 WMMA builtin): one thread per point, scalar loop.
__global__ __launch_bounds__(256) void splat_fallback(
    const float* __restrict__ featP, const float* __restrict__ coefP,
    const float* __restrict__ vval, float* __restrict__ out)
{
    int p = blockIdx.x * 256 + threadIdx.x;
    if (p >= MP) return;
    float f[10];
#pragma unroll
    for (int k = 0; k < 10; ++k) f[k] = featP[(k>>1)*(2*MP) + 2*p + (k&1)];
    float num = 0.f, den = 0.f;
    for (int g = 0; g < NG; ++g) {
        float q = 0.f;
#pragma unroll
        for (int k = 0; k < 10; ++k) q += f[k] * coefP[(k>>1)*(2*NG) + 2*g + (k&1)];
        float alpha = fast_exp2(q);
        num += alpha * vval[g];
        den += alpha;
    }
    out[p] = (den > 1e-8f) ? num : BG_VAL;
}
#endif

// ---------------------------------------------------------------------------
extern "C" void kernel_launch(void* const* d_in, const int* in_sizes, int n_in,
                              void* d_out, int out_size, void* d_ws, size_t ws_size,
                              hipStream_t stream)
{
    const float* x    = (const float*)d_in[0];   // [MP,3]
    const float* xyz  = (const float*)d_in[1];   // [NG,3]
    const float* wraw = (const float*)d_in[2];   // [NG,1]
    const float* sraw = (const float*)d_in[3];   // [NG,3]
    const float* rraw = (const float*)d_in[4];   // [NG,4]
    const float* vraw = (const float*)d_in[5];   // [NG,1]

    float* ws    = (float*)d_ws;
    float* coefP = ws;                    // KF*NG floats
    float* vval  = coefP + KF*NG;         // NG floats
    float* featP = vval + NG;             // KF*MP floats
    float* numP  = featP + KF*MP;         // GSPLIT*MP floats
    float* denP  = numP + GSPLIT*MP;      // GSPLIT*MP floats  (total ~2.4 MB)

    precompute_gauss<<<NG/256, 256, 0, stream>>>(xyz, wraw, sraw, rraw, vraw,
                                                 coefP, vval);
    precompute_points<<<MP/256, 256, 0, stream>>>(x, featP);

#if HAVE_WMMA_F32X4
    splat_main<<<dim3(MP/16, GSPLIT), 256, 0, stream>>>(featP, coefP, vval,
                                                        numP, denP);
    finalize<<<MP/256, 256, 0, stream>>>(numP, denP, (float*)d_out);
#else
    splat_fallback<<<MP/256, 256, 0, stream>>>(featP, coefP, vval, (float*)d_out);
#endif
}